// YoloLoss_65798898974851
// MI455X (gfx1250) — compile-verified
//
#include <hip/hip_runtime.h>
#include <hip/hip_bf16.h>

// ---------------------------------------------------------------------------
// YOLO-v1 loss for MI455X (gfx1250). Pure bandwidth problem (~174 MB in,
// 20 B out, ~0.14 GFLOP): ~7.5 us at 23.3 TB/s HBM; working set fits in the
// 192 MB global L2, so replays run at L2 speed -> L0 line-touch efficiency
// dominates. Strategy:
//   * block-cooperative staging of pred/tcls/tbox into LDS with fully
//     coalesced b128 loads (async GLOBAL_LOAD_ASYNC_TO_LDS_B128 when the
//     toolchain exposes it, plain coalesced float4 copies otherwise)
//   * per-thread consume from LDS, cheap VALU per cell
//   * wave32 reduction with V_WMMA_F32_16X16X4_F32 (ones B-matrix): one wmma
//     + 8 lane-local adds + 1 shfl_xor(16) = full 32-lane f32 sum.
// ---------------------------------------------------------------------------

#define S_GRID 14
#define N_BATCH 4096
#define N_CELLS (N_BATCH * S_GRID * S_GRID)   // 802,816
#define MAIN_BLOCKS 784
#define MAIN_THREADS 256
#define CELLS_PER_ITER 256
#define ITERS 4                               // 784 * 256 * 4 == N_CELLS exactly
#define FIN_THREADS 128                       // 4 full waves, one per term

typedef float v2f __attribute__((ext_vector_type(2)));
typedef float v8f __attribute__((ext_vector_type(8)));

#if __has_builtin(__builtin_amdgcn_global_load_async_to_lds_b128)
#define HAVE_ASYNC_LDS 1
#else
#define HAVE_ASYNC_LDS 0
#endif

// Exact builtin parameter types (from hipcc diagnostic):
//   param0: int __attribute__((vector_size(16))) * in AS(1)  (global src)
//   param1: LDS destination (AS(3)), same vector type
typedef int v4i_g __attribute__((vector_size(16)));
typedef __attribute__((address_space(1))) v4i_g g_v4i;
typedef __attribute__((address_space(3))) v4i_g l_v4i;

// Coalesced b128 staging: global -> LDS. Async path writes LDS directly
// (no VGPR round-trip, tracked with ASYNCcnt); fallback is a coalesced copy.
__device__ __forceinline__ void stage_f4(const float4* __restrict__ g,
                                         float4* s, int count, int t) {
#if HAVE_ASYNC_LDS
    for (int i = t; i < count; i += MAIN_THREADS) {
        __builtin_amdgcn_global_load_async_to_lds_b128(
            (g_v4i*)(g + i), (l_v4i*)(s + i), /*offset=*/0, /*cpol=*/0);
    }
#else
    for (int i = t; i < count; i += MAIN_THREADS) s[i] = g[i];
#endif
}

__device__ __forceinline__ void stage_wait() {
#if HAVE_ASYNC_LDS
#if __has_builtin(__builtin_amdgcn_s_wait_asynccnt)
    __builtin_amdgcn_s_wait_asynccnt(0);
#else
    asm volatile("s_wait_asynccnt 0x0" ::: "memory");
#endif
#endif
    __syncthreads();
}

// Full wave32 sum via V_WMMA_F32_16X16X4_F32.
// A (16x4): lane m%16 supplies K=0/1 (lanes 0-15) or K=2/3 (lanes 16-31);
// with a[1]=0 and B=ones: D[m][n] = v_m + v_{m+16}. Lane-local sum of the
// 8 D regs gives rows 0-7 (lanes 0-15) / rows 8-15 (lanes 16-31); one
// shfl_xor(16) add completes the full 32-lane f32 sum.
__device__ __forceinline__ float wave_sum_wmma(float v) {
    v2f a; a[0] = v;    a[1] = 0.0f;
    v2f b; b[0] = 1.0f; b[1] = 1.0f;
    v8f c = {};
    v8f d = __builtin_amdgcn_wmma_f32_16x16x4_f32(
        false, a, false, b, (short)0, c, false, false);
    float s = ((d[0] + d[1]) + (d[2] + d[3])) + ((d[4] + d[5]) + (d[6] + d[7]));
    s += __shfl_xor(s, 16, 32);
    return s;
}

struct BoxXYXY { float x0, y0, x1, y1; };

__device__ __forceinline__ BoxXYXY xywh2xyxy(float x, float y, float w, float h) {
    const float inv_s = 1.0f / (float)S_GRID;
    BoxXYXY b;
    b.x0 = x * inv_s - 0.5f * w;
    b.y0 = y * inv_s - 0.5f * h;
    b.x1 = x * inv_s + 0.5f * w;
    b.y1 = y * inv_s + 0.5f * h;
    return b;
}

__device__ __forceinline__ float iou(const BoxXYXY& a, const BoxXYXY& b) {
    float lx = fmaxf(a.x0, b.x0), ly = fmaxf(a.y0, b.y0);
    float rx = fminf(a.x1, b.x1), ry = fminf(a.y1, b.y1);
    float w = fmaxf(rx - lx, 0.0f), h = fmaxf(ry - ly, 0.0f);
    float inter = w * h;
    float area_a = (a.x1 - a.x0) * (a.y1 - a.y0);
    float area_b = (b.x1 - b.x0) * (b.y1 - b.y0);
    return inter / (area_a + area_b - inter);
}

__global__ __launch_bounds__(MAIN_THREADS)
void yolo_main_kernel(const float* __restrict__ pred,
                      const float* __restrict__ tbox,
                      const float* __restrict__ tcls,
                      const int*   __restrict__ hmap,
                      float* __restrict__ block_part) {
    __shared__ float s_pred[CELLS_PER_ITER * 30];   // 30 KB
    __shared__ float s_tcls[CELLS_PER_ITER * 20];   // 20 KB
    __shared__ float s_tbox[CELLS_PER_ITER * 4];    //  4 KB
    __shared__ float s_red[MAIN_THREADS / 32][4];

    const int t = threadIdx.x;
    float acc_reg = 0.0f, acc_cont = 0.0f, acc_noobj = 0.0f, acc_cls = 0.0f;

    for (int it = 0; it < ITERS; ++it) {
        const int cellBase = (blockIdx.x * ITERS + it) * CELLS_PER_ITER;

        // ---- coalesced b128 staging into LDS (chunk bases 16B aligned:
        //      cellBase is a multiple of 256) ----
        stage_f4(reinterpret_cast<const float4*>(pred + (size_t)cellBase * 30),
                 reinterpret_cast<float4*>(s_pred), CELLS_PER_ITER * 30 / 4, t);
        stage_f4(reinterpret_cast<const float4*>(tcls + (size_t)cellBase * 20),
                 reinterpret_cast<float4*>(s_tcls), CELLS_PER_ITER * 20 / 4, t);
        stage_f4(reinterpret_cast<const float4*>(tbox + (size_t)cellBase * 4),
                 reinterpret_cast<float4*>(s_tbox), CELLS_PER_ITER * 4 / 4, t);
        const float m = hmap[cellBase + t] ? 1.0f : 0.0f;   // already coalesced
        stage_wait();

        // ---- consume: thread t owns staged cell t ----
        float pv[30];
        #pragma unroll
        for (int k = 0; k < 30; ++k) pv[k] = s_pred[t * 30 + k];
        float tc[20];
        #pragma unroll
        for (int k = 0; k < 20; ++k) tc[k] = s_tcls[t * 20 + k];
        float tbx = s_tbox[t * 4 + 0], tby = s_tbox[t * 4 + 1];
        float tbw = s_tbox[t * 4 + 2], tbh = s_tbox[t * 4 + 3];

        // classification loss
        float cl = 0.0f;
        #pragma unroll
        for (int k = 0; k < 20; ++k) {
            float d = pv[10 + k] - tc[k];
            cl += d * d;
        }
        acc_cls += m * cl;

        // no-object confidence loss
        acc_noobj += (1.0f - m) * (pv[4] * pv[4] + pv[9] * pv[9]);

        // responsible-box selection via IoU
        BoxXYXY tgt = xywh2xyxy(tbx, tby, tbw, tbh);
        BoxXYXY b1  = xywh2xyxy(pv[0], pv[1], pv[2], pv[3]);
        BoxXYXY b2  = xywh2xyxy(pv[5], pv[6], pv[7], pv[8]);
        float iou1 = iou(tgt, b1);
        float iou2 = iou(tgt, b2);
        bool use1 = iou1 > iou2;
        float best_iou = use1 ? iou1 : iou2;
        float bx = use1 ? pv[0] : pv[5];
        float by = use1 ? pv[1] : pv[6];
        float bw = use1 ? pv[2] : pv[7];
        float bh = use1 ? pv[3] : pv[8];
        float bc = use1 ? pv[4] : pv[9];

        float dx = bx - tbx, dy = by - tby;
        float sw = sqrtf(bw) - sqrtf(tbw);
        float sh = sqrtf(bh) - sqrtf(tbh);
        acc_reg  += m * ((dx * dx + dy * dy) + (sw * sw + sh * sh));
        float dcont = bc - best_iou;
        acc_cont += m * (dcont * dcont);

        __syncthreads();   // protect LDS before next iteration's staging
    }

    // ---- wave32 WMMA reduction, then 8-wave LDS combine ----
    const int wave = t >> 5;
    const int lane = t & 31;
    float r0 = wave_sum_wmma(acc_reg);
    float r1 = wave_sum_wmma(acc_cont);
    float r2 = wave_sum_wmma(acc_noobj);
    float r3 = wave_sum_wmma(acc_cls);
    if (lane == 0) {
        s_red[wave][0] = r0; s_red[wave][1] = r1;
        s_red[wave][2] = r2; s_red[wave][3] = r3;
    }
    __syncthreads();
    if (t < 4) {
        float s = 0.0f;
        #pragma unroll
        for (int w = 0; w < MAIN_THREADS / 32; ++w) s += s_red[w][t];
        block_part[(size_t)blockIdx.x * 4 + t] = s;
    }
}

__global__ __launch_bounds__(FIN_THREADS)
void yolo_finalize_kernel(const float* __restrict__ block_part,
                          float* __restrict__ out) {
    const int wave = threadIdx.x >> 5;   // term: 0=reg 1=cont 2=noobj 3=cls
    const int lane = threadIdx.x & 31;

    float s = 0.0f;
    for (int i = lane; i < MAIN_BLOCKS; i += 32)
        s += block_part[(size_t)i * 4 + wave];
    s = wave_sum_wmma(s);

    __shared__ float terms[4];
    if (lane == 0) terms[wave] = s;
    __syncthreads();

    if (threadIdx.x == 0) {
        float reg = terms[0], cont = terms[1], noobj = terms[2], cls = terms[3];
        out[0] = 5.0f * reg + cont + 0.5f * noobj + cls;  // L_COORD=5, L_NOOBJ=0.5
        out[1] = reg;
        out[2] = cont;
        out[3] = noobj;
        out[4] = cls;
    }
}

extern "C" void kernel_launch(void* const* d_in, const int* in_sizes, int n_in,
                              void* d_out, int out_size, void* d_ws, size_t ws_size,
                              hipStream_t stream) {
    (void)in_sizes; (void)n_in; (void)out_size; (void)ws_size;
    const float* pred = (const float*)d_in[0];   // (4096,14,14,30) f32
    const float* tbox = (const float*)d_in[1];   // (4096,14,14,4)  f32
    const float* tcls = (const float*)d_in[2];   // (4096,14,14,20) f32
    const int*   hmap = (const int*)  d_in[3];   // (4096,14,14)    bool->int
    float* out = (float*)d_out;                  // 5 f32
    float* part = (float*)d_ws;                  // MAIN_BLOCKS*4 f32 = 12.5 KB

    yolo_main_kernel<<<MAIN_BLOCKS, MAIN_THREADS, 0, stream>>>(pred, tbox, tcls, hmap, part);
    yolo_finalize_kernel<<<1, FIN_THREADS, 0, stream>>>(part, out);
}